// HybridLoss_8048768713313
// MI455X (gfx1250) — compile-verified
//
#include <hip/hip_runtime.h>
#include <hip/hip_bf16.h>

// ---------------------------------------------------------------------------
// HybridLoss for (B=8, T=20, C=1, H=480, W=480) fp32 tensors.
// Phase 1: streaming NT pass, 22 statistics per (b,t) slice, block partials
//          (rows padded to 32 cols / 48 rows with zeros for branch-free WMMA).
// Phase 2: WMMA f32 16x16x4 (A = ones) -> exact fp32 column sums per slice.
// Phase 3: one wave combines over b, applies weights, writes the scalar.
// ---------------------------------------------------------------------------

typedef __attribute__((ext_vector_type(2))) float v2f;
typedef __attribute__((ext_vector_type(4))) float v4f;
typedef __attribute__((ext_vector_type(8))) float v8f;

#define BB 8
#define TT 20
#define HW 230400            // 480*480
#define NB 45                // real data blocks per (b,t) slice
#define NBP 48               // padded K dimension (multiple of 4 for WMMA)
#define ELEMS_PER_BLOCK 5120 // 230400 / 45  (= 5 float4 per thread @ 256 thr)
#define QS 22                // number of statistics per slice
#define QSP 32               // padded row width (two 16-wide WMMA tiles)
#define NSLICE (BB * TT)     // 160

// thresholds: log(x+1)/log(31) for x in {0.1,1,2,5,8}
__device__ __constant__ float c_TH[5] = {
    0.02775517f, 0.20184909f, 0.31992332f, 0.52177215f, 0.63984664f};
__device__ __constant__ float c_CSIW[5] = {0.1f, 0.1f, 0.2f, 0.25f, 0.35f};
__device__ __constant__ float c_TW[20] = {
    0.0075f, 0.02f, 0.03f, 0.04f, 0.05f, 0.06f, 0.07f, 0.08f, 0.09f, 0.1f,
    0.09f,  0.08f, 0.07f, 0.06f, 0.05f, 0.04f, 0.03f, 0.02f, 0.0075f, 0.005f};

// ---------------------------------------------------------------------------
// Phase 1: grid = NSLICE*NBP blocks x 256 threads.  Blocks with blk < NB
// consume 5120 contiguous elements of one (b,t) slice with non-temporal
// 128-bit loads; pad blocks (blk >= NB) emit zero rows so phase 2 needs no
// bound checks.  Output: ws1[blockIdx.x][0..31] (cols 22..31 zero).
// ---------------------------------------------------------------------------
__global__ __launch_bounds__(256) void hl_phase1(
    const float* __restrict__ pred, const float* __restrict__ targ,
    const float* __restrict__ mask, float* __restrict__ ws1)
{
    const int slice = blockIdx.x / NBP;
    const int blk   = blockIdx.x % NBP;
    const int tid   = threadIdx.x;

    float acc[QS];
#pragma unroll
    for (int q = 0; q < QS; ++q) acc[q] = 0.0f;

    if (blk < NB) {
        const long base = (long)slice * HW + (long)blk * ELEMS_PER_BLOCK;

        // hoisted K_th = exp(30*th)
        float K[5];
#pragma unroll
        for (int i = 0; i < 5; ++i) K[i] = __expf(30.0f * c_TH[i]);

#pragma unroll
        for (int it = 0; it < 5; ++it) {
            const long off = base + (long)(it * 256 + tid) * 4;
            const v4f p4 = __builtin_nontemporal_load(
                reinterpret_cast<const v4f*>(pred + off));
            const v4f t4 = __builtin_nontemporal_load(
                reinterpret_cast<const v4f*>(targ + off));
            const v4f m4 = __builtin_nontemporal_load(
                reinterpret_cast<const v4f*>(mask + off));
#pragma unroll
            for (int e = 0; e < 4; ++e) {
                float p = fminf(fmaxf(p4[e], 0.0f), 1.0f);
                const float t = t4[e];
                const float m = m4[e];
                // weighted MAE
                acc[0] += fabsf(p - t) * fmaf(10.0f * t, t, 1.0f) * m;
                acc[1] += m;
                // soft CSI terms: sigmoid((p-th)*30) = E / (E + exp(30*th))
                const float E = __expf(30.0f * p);
#pragma unroll
                for (int i = 0; i < 5; ++i) {
                    const float s  = __fdividef(E, E + K[i]);
                    const float ps = s * m;
                    const float ts = (t > c_TH[i]) ? m : 0.0f;
                    acc[2 + i]  += ps;
                    acc[7 + i]  += ts;
                    acc[12 + i] += ps * ts;
                }
                // correlation moments on (p*m, t*m)
                const float pm = p * m, tm = t * m;
                acc[17] += pm;
                acc[18] += tm;
                acc[19] += pm * tm;
                acc[20] += pm * pm;
                acc[21] += tm * tm;
            }
        }
    }

    // wave32 butterfly reduction
#pragma unroll
    for (int q = 0; q < QS; ++q) {
#pragma unroll
        for (int d = 1; d < 32; d <<= 1)
            acc[q] += __shfl_xor(acc[q], d, 32);
    }

    __shared__ float sred[8][QSP];
    const int wave = tid >> 5, lane = tid & 31;
    if (lane == 0) {
#pragma unroll
        for (int q = 0; q < QSP; ++q)
            sred[wave][q] = (q < QS) ? acc[q] : 0.0f;
    }
    __syncthreads();

    if (tid < QSP) {
        float s = 0.0f;
#pragma unroll
        for (int w = 0; w < 8; ++w) s += sred[w][tid];
        ws1[(long)blockIdx.x * QSP + tid] = s;
    }
}

// ---------------------------------------------------------------------------
// Phase 2: grid = NSLICE blocks x 32 threads (one wave).  Sum the 48 padded
// rows of 32 partials per slice with V_WMMA_F32_16X16X4_F32, A = all ones:
//   D[m][n] = sum_k B[k][n]  (exact fp32 column sums).
// B layout (analogous to the documented 32-bit A 16x4 layout):
//   VGPR v, lane L -> B[k = (L<16 ? v : v+2)][n = L%16].
// All loads are branch-free thanks to the zero padding written by phase 1.
// ---------------------------------------------------------------------------
__global__ __launch_bounds__(32) void hl_phase2(
    const float* __restrict__ ws1, float* __restrict__ ws2)
{
    const int slice = blockIdx.x;
    const int lane  = threadIdx.x;
    const float* base = ws1 + (long)slice * NBP * QSP;

    const int n     = lane & 15;
    const int khalf = (lane < 16) ? 0 : 2;

    v2f a; a[0] = 1.0f; a[1] = 1.0f;
    v8f c0 = {}, c1 = {};

#pragma unroll
    for (int kb = 0; kb < NBP; kb += 4) {
        const int k0 = kb + khalf;
        const int k1 = k0 + 1;
        v2f b0, b1;
        b0[0] = base[k0 * QSP + n];
        b0[1] = base[k1 * QSP + n];
        b1[0] = base[k0 * QSP + 16 + n];
        b1[1] = base[k1 * QSP + 16 + n];
        c0 = __builtin_amdgcn_wmma_f32_16x16x4_f32(
            false, a, false, b0, (short)0, c0, false, false);
        c1 = __builtin_amdgcn_wmma_f32_16x16x4_f32(
            false, a, false, b1, (short)0, c1, false, false);
    }

    // D VGPR0, lanes 0..15 hold row M=0 -> per-column (per-statistic) sums
    if (lane < 16) {
        ws2[(long)slice * QSP + lane]      = c0[0];
        ws2[(long)slice * QSP + 16 + lane] = c1[0];
    }
}

// ---------------------------------------------------------------------------
// Phase 3: single wave.  Lane t (< 20) combines over b, forms frame loss,
// applies time weights, butterfly-reduces over T, writes the scalar.
// ---------------------------------------------------------------------------
__global__ __launch_bounds__(32) void hl_phase3(
    const float* __restrict__ ws2, float* __restrict__ out)
{
    const int t = threadIdx.x;
    float val = 0.0f;
    if (t < TT) {
        float maes = 0.0f, ms = 0.0f, rsum = 0.0f;
        float csum[5] = {0.0f, 0.0f, 0.0f, 0.0f, 0.0f};
        const float N = (float)HW;
        for (int b = 0; b < BB; ++b) {
            const float* s = ws2 + (long)(b * TT + t) * QSP;
            maes += s[0];
            ms   += s[1];
#pragma unroll
            for (int i = 0; i < 5; ++i) {
                const float inter = s[12 + i];
                const float uni   = s[2 + i] + s[7 + i] - inter;
                csum[i] += (inter + 1e-6f) / (uni + 1e-6f);
            }
            const float num = s[19] - s[17] * s[18] / N;
            const float vp  = s[20] - s[17] * s[17] / N;
            const float vt  = s[21] - s[18] * s[18] / N;
            rsum += num / (sqrtf(vp * vt) + 1e-6f);
        }
        const float l_mae = maes / (ms + 1e-8f);
        float l_csi = 0.0f;
#pragma unroll
        for (int i = 0; i < 5; ++i)
            l_csi += c_CSIW[i] * (1.0f - csum[i] * 0.125f);
        const float l_corr = 1.0f - rsum * 0.125f;
        val = (10.0f * l_mae + l_csi + l_corr) * c_TW[t] * 20.0f;
    }
#pragma unroll
    for (int d = 1; d < 32; d <<= 1) val += __shfl_xor(val, d, 32);
    if (t == 0) out[0] = val;
}

// ---------------------------------------------------------------------------
extern "C" void kernel_launch(void* const* d_in, const int* in_sizes, int n_in,
                              void* d_out, int out_size, void* d_ws, size_t ws_size,
                              hipStream_t stream)
{
    const float* pred = (const float*)d_in[0];
    const float* targ = (const float*)d_in[1];
    const float* mask = (const float*)d_in[2];
    float* out = (float*)d_out;

    float* ws1 = (float*)d_ws;                        // NSLICE*NBP*QSP floats
    float* ws2 = ws1 + (long)NSLICE * NBP * QSP;      // NSLICE*QSP floats

    hl_phase1<<<NSLICE * NBP, 256, 0, stream>>>(pred, targ, mask, ws1);
    hl_phase2<<<NSLICE, 32, 0, stream>>>(ws1, ws2);
    hl_phase3<<<1, 32, 0, stream>>>(ws2, out);
    (void)in_sizes; (void)n_in; (void)out_size; (void)ws_size;
}